// IDWT_2D_16449724745581
// MI455X (gfx1250) — compile-verified
//
#include <hip/hip_runtime.h>
#include <hip/hip_bf16.h>
#include <stdint.h>

typedef float v4f __attribute__((ext_vector_type(4)));

// IDWT (Haar, 2x2, stride 2, non-overlapping) — pure HBM streaming kernel.
// Input bands: (8,64,256,256) f32 each. Output: (8,64,512,512) f32.
// Per thread: one float4 quad per band -> async-copy to LDS (NT hint) ->
// butterfly -> four NT b128 stores (two per output row).
__global__ __launch_bounds__(256) void idwt_haar_async_kernel(
    const float* __restrict__ LL, const float* __restrict__ LH,
    const float* __restrict__ HL, const float* __restrict__ HH,
    float* __restrict__ out, long long nquads) {
  // 4 bands x 256 threads x 4 floats = 16 KB LDS per block.
  __shared__ __align__(16) float sm[4][1024];

  long long q = (long long)blockIdx.x * blockDim.x + threadIdx.x;
  if (q >= nquads) return;

  long long n = q << 2;                       // element index into a band
  unsigned row = (unsigned)(n >> 8);          // global input row (w = 256)
  unsigned col = (unsigned)(n & 255);         // quad start column
  unsigned img = row >> 8;                    // image index (b*C + c), h = 256
  unsigned i   = row & 255;                   // row within image
  // out image = 512*512; out row stride = 512
  long long obase = (long long)img * 262144 + (long long)i * 1024 + (long long)(col * 2);

  unsigned t4 = threadIdx.x * 4u;
  // Flat shared pointer = {aperture_hi32, lds_offset}; low 32 bits = LDS byte addr.
  unsigned lds0 = (unsigned)(uintptr_t)&sm[0][t4];
  unsigned lds1 = (unsigned)(uintptr_t)&sm[1][t4];
  unsigned lds2 = (unsigned)(uintptr_t)&sm[2][t4];
  unsigned lds3 = (unsigned)(uintptr_t)&sm[3][t4];

  unsigned long long a0 = (unsigned long long)(uintptr_t)(LL + n);
  unsigned long long a1 = (unsigned long long)(uintptr_t)(LH + n);
  unsigned long long a2 = (unsigned long long)(uintptr_t)(HL + n);
  unsigned long long a3 = (unsigned long long)(uintptr_t)(HH + n);

  // CDNA5 async global->LDS copies (ASYNCcnt-tracked), GV addressing mode.
  // Inputs are read-once (512 MiB >> 192 MB L2): non-temporal hint.
  asm volatile("global_load_async_to_lds_b128 %0, %1, off th:TH_LOAD_NT"
               :: "v"(lds0), "v"(a0) : "memory");
  asm volatile("global_load_async_to_lds_b128 %0, %1, off th:TH_LOAD_NT"
               :: "v"(lds1), "v"(a1) : "memory");
  asm volatile("global_load_async_to_lds_b128 %0, %1, off th:TH_LOAD_NT"
               :: "v"(lds2), "v"(a2) : "memory");
  asm volatile("global_load_async_to_lds_b128 %0, %1, off th:TH_LOAD_NT"
               :: "v"(lds3), "v"(a3) : "memory");
  asm volatile("s_wait_asynccnt 0" ::: "memory");

  v4f ll = *(const v4f*)&sm[0][t4];
  v4f lh = *(const v4f*)&sm[1][t4];
  v4f hl = *(const v4f*)&sm[2][t4];
  v4f hh = *(const v4f*)&sm[3][t4];

  // Haar butterfly (weights are +-0.5)
  v4f s1 = ll + lh;   // row p=0 partial
  v4f s2 = ll - lh;   // row p=1 partial
  v4f t1 = hl + hh;
  v4f t2 = hl - hh;
  v4f e00 = 0.5f * (s1 + t1);  // out[2i  ][2j  ]
  v4f e01 = 0.5f * (s1 - t1);  // out[2i  ][2j+1]
  v4f e10 = 0.5f * (s2 + t2);  // out[2i+1][2j  ]
  v4f e11 = 0.5f * (s2 - t2);  // out[2i+1][2j+1]

  // Interleave the 4 input columns into 8 contiguous output columns per row.
  v4f o00 = {e00.x, e01.x, e00.y, e01.y};
  v4f o01 = {e00.z, e01.z, e00.w, e01.w};
  v4f o10 = {e10.x, e11.x, e10.y, e11.y};
  v4f o11 = {e10.z, e11.z, e10.w, e11.w};

  // Write-once 512 MiB output: non-temporal b128 stores (don't thrash 192MB L2).
  __builtin_nontemporal_store(o00, (v4f*)(out + obase));
  __builtin_nontemporal_store(o01, (v4f*)(out + obase + 4));
  __builtin_nontemporal_store(o10, (v4f*)(out + obase + 512));
  __builtin_nontemporal_store(o11, (v4f*)(out + obase + 516));
}

extern "C" void kernel_launch(void* const* d_in, const int* in_sizes, int n_in,
                              void* d_out, int out_size, void* d_ws, size_t ws_size,
                              hipStream_t stream) {
  const float* LL = (const float*)d_in[0];
  const float* LH = (const float*)d_in[1];
  const float* HL = (const float*)d_in[2];
  const float* HH = (const float*)d_in[3];
  float* out = (float*)d_out;

  long long total = (long long)in_sizes[0];     // 8*64*256*256 = 33,554,432
  long long nquads = total >> 2;                // float4 quads per band
  const int threads = 256;
  unsigned blocks = (unsigned)((nquads + threads - 1) / threads);

  idwt_haar_async_kernel<<<blocks, threads, 0, stream>>>(LL, LH, HL, HH, out, nquads);
}